// MakeLSTM_6356551598839
// MI455X (gfx1250) — compile-verified
//
#include <hip/hip_runtime.h>
#include <math.h>

typedef float v2f __attribute__((ext_vector_type(2)));
typedef float v8f __attribute__((ext_vector_type(8)));

#define D_MODEL 1792
#define BB      256
#define I_DIM   2048      // inner dim
#define G_DIM   8192      // 4*I

#define WMMA_F32(a, b, c) \
    __builtin_amdgcn_wmma_f32_16x16x4_f32(false, (a), false, (b), (short)0, (c), false, false)

// GEMV for one LSTM cell: g = Wih@xt + bih + Whh@h + bhh
// xt = concat(xA[0:KA], xB[0:Kih-KA]); grid.x = G_DIM/16, block = 256 (8 waves).
// Waves 0-3 cover Whh (K=2048), waves 4-7 cover Wih (K=Kih). Each wave owns a
// contiguous K-chunk and accumulates 16 output rows with V_WMMA_F32_16X16X4_F32
// (B = x broadcast to all 16 columns -> every D column holds the row sums).
// Inner loop: double-buffered blocks of 8 NT b64 loads to keep HBM saturated.
__global__ __launch_bounds__(256)
void lstm_gemv(const float* __restrict__ Wih, int Kih,
               const float* __restrict__ xA, int KA, const float* __restrict__ xB,
               const float* __restrict__ Whh,
               const float* __restrict__ hvec,
               const float* __restrict__ bih, const float* __restrict__ bhh,
               float* __restrict__ g)
{
    __shared__ float sx[I_DIM];     // xt
    __shared__ float sh[I_DIM];     // h
    __shared__ float red[8 * 16];   // per-wave partial row sums

    const int tid = threadIdx.x;
    for (int idx = tid; idx < Kih; idx += 256)
        sx[idx] = (idx < KA) ? xA[idx] : xB[idx - KA];
    for (int idx = tid; idx < I_DIM; idx += 256)
        sh[idx] = hvec[idx];
    __syncthreads();

    const int wave = tid >> 5;
    const int lane = tid & 31;
    const int m0   = blockIdx.x * 16;
    const int mrow = m0 + (lane & 15);
    const int cb   = (lane >> 4) << 1;   // K sub-offset per ISA A/B layout

    const float* Wp; const float* xv; int ldk, kChunk, kBeg;
    if (wave < 4) { Wp = Whh; xv = sh; ldk = I_DIM; kChunk = I_DIM >> 2; kBeg = wave * kChunk; }
    else          { Wp = Wih; xv = sx; ldk = Kih;  kChunk = Kih   >> 2; kBeg = (wave - 4) * kChunk; }

    const float* row = Wp + (size_t)mrow * (size_t)ldk + kBeg + cb;  // 8B aligned
    const float* xp  = xv + kBeg + cb;

    // 32 K-steps per block; chunk is 512 or 448 -> nBlk = 16 or 14 (always even).
    // Wave-uniform by construction; readfirstlane forces scalar loop control so
    // no EXEC masking surrounds the WMMAs.
    const int nBlk = __builtin_amdgcn_readfirstlane(kChunk >> 5);

    v8f acc = {};
    v2f a0[8], a1[8];

    #pragma unroll
    for (int u = 0; u < 8; ++u)
        a0[u] = __builtin_nontemporal_load((const v2f*)(row + 4 * u));

    for (int it = 0; it < nBlk; it += 2) {
        const float* r1 = row + (size_t)(it + 1) * 32;
        #pragma unroll
        for (int u = 0; u < 8; ++u)          // prefetch odd block
            a1[u] = __builtin_nontemporal_load((const v2f*)(r1 + 4 * u));

        const float* xk0 = xp + (size_t)it * 32;
        #pragma unroll
        for (int u = 0; u < 8; ++u) {        // consume even block
            v2f b; b.x = xk0[4 * u]; b.y = xk0[4 * u + 1];
            acc = WMMA_F32(a0[u], b, acc);
        }

        if (it + 2 < nBlk) {
            const float* r2 = row + (size_t)(it + 2) * 32;
            #pragma unroll
            for (int u = 0; u < 8; ++u)      // prefetch next even block
                a0[u] = __builtin_nontemporal_load((const v2f*)(r2 + 4 * u));
        }

        const float* xk1 = xp + (size_t)(it + 1) * 32;
        #pragma unroll
        for (int u = 0; u < 8; ++u) {        // consume odd block
            v2f b; b.x = xk1[4 * u]; b.y = xk1[4 * u + 1];
            acc = WMMA_F32(a1[u], b, acc);
        }
    }

    // D layout: VGPR v, lanes 0-15 -> M=v, lanes 16-31 -> M=v+8; all N identical.
    if ((lane & 15) == 0) {
        const int half = lane >> 4;
        #pragma unroll
        for (int v = 0; v < 8; ++v)
            red[wave * 16 + half * 8 + v] = acc[v];
    }
    __syncthreads();

    if (tid < 16) {                       // fixed-order reduction: deterministic
        float s = 0.f;
        #pragma unroll
        for (int w = 0; w < 8; ++w) s += red[w * 16 + tid];
        const int m = m0 + tid;
        g[m] = s + bih[m] + bhh[m];
    }
}

// Elementwise LSTM gate update (i,f,g,o quarters of g); optionally emit h tail.
__global__ __launch_bounds__(256)
void lstm_pw(const float* __restrict__ g, float* __restrict__ h,
             float* __restrict__ c, float* __restrict__ out)
{
    const int j = blockIdx.x * blockDim.x + threadIdx.x;
    if (j >= I_DIM) return;
    const float gi = g[j];
    const float gf = g[I_DIM + j];
    const float gg = g[2 * I_DIM + j];
    const float go = g[3 * I_DIM + j];
    const float si = 1.f / (1.f + expf(-gi));
    const float sf = 1.f / (1.f + expf(-gf));
    const float so = 1.f / (1.f + expf(-go));
    const float nc = sf * c[j] + si * tanhf(gg);
    const float nh = so * tanhf(nc);
    c[j] = nc;
    h[j] = nh;
    if (out != nullptr && j >= I_DIM - BB) out[j - (I_DIM - BB)] = nh;
}

__global__ void lstm_init(float* h, float* c)
{
    const int j = blockIdx.x * blockDim.x + threadIdx.x;
    if (j < I_DIM) { h[j] = 0.f; c[j] = 0.f; }
}

extern "C" void kernel_launch(void* const* d_in, const int* in_sizes, int n_in,
                              void* d_out, int out_size, void* d_ws, size_t ws_size,
                              hipStream_t stream)
{
    const float* x    = (const float*)d_in[0];  // (6, 1792)
    const float* y    = (const float*)d_in[1];  // (5, 256)
    const float* Wih5 = (const float*)d_in[2];  // (5, 8192, 2048)
    const float* Whh5 = (const float*)d_in[3];  // (5, 8192, 2048)
    const float* bih5 = (const float*)d_in[4];  // (5, 8192)
    const float* bhh5 = (const float*)d_in[5];  // (5, 8192)
    const float* WihF = (const float*)d_in[6];  // (8192, 1792)
    const float* WhhF = (const float*)d_in[7];  // (8192, 2048)
    const float* bihF = (const float*)d_in[8];  // (8192,)
    const float* bhhF = (const float*)d_in[9];  // (8192,)
    float* out = (float*)d_out;                 // 256 floats

    float* ws = (float*)d_ws;
    float* h  = ws;
    float* c  = ws + I_DIM;
    float* g  = ws + 2 * I_DIM;                 // 8192 floats

    lstm_init<<<I_DIM / 256, 256, 0, stream>>>(h, c);

    const size_t Wstride = (size_t)G_DIM * (size_t)I_DIM;
    for (int t = 0; t < 5; ++t) {
        lstm_gemv<<<G_DIM / 16, 256, 0, stream>>>(
            Wih5 + (size_t)t * Wstride, I_DIM,
            x + (size_t)t * D_MODEL, D_MODEL, y + (size_t)t * BB,
            Whh5 + (size_t)t * Wstride, h,
            bih5 + (size_t)t * G_DIM, bhh5 + (size_t)t * G_DIM, g);
        lstm_pw<<<I_DIM / 256, 256, 0, stream>>>(g, h, c, nullptr);
    }
    // Final cell: input is x[5] only (K = 1792)
    lstm_gemv<<<G_DIM / 16, 256, 0, stream>>>(
        WihF, D_MODEL,
        x + 5 * (size_t)D_MODEL, D_MODEL, y /*unused: KA==Kih*/,
        WhhF, h, bihF, bhhF, g);
    lstm_pw<<<I_DIM / 256, 256, 0, stream>>>(g, h, c, out);
}